// TibGroupLasso_39685497815125
// MI455X (gfx1250) — compile-verified
//
#include <hip/hip_runtime.h>

#define B_ROWS 4096
#define F_COLS 16384
#define G_GRP  512
#define S_GRP  32

typedef __attribute__((ext_vector_type(2))) float v2f;
typedef __attribute__((ext_vector_type(8))) float v8f;

// ---------------------------------------------------------------------------
// Kernel 0: zero the scatter target (wflat, F floats in d_ws) and d_out.
// ---------------------------------------------------------------------------
__global__ void tib_zero(float* __restrict__ wflat, float* __restrict__ out) {
    int i = blockIdx.x * blockDim.x + threadIdx.x;
    if (i < F_COLS) wflat[i] = 0.0f;
    if (i < B_ROWS) out[i]   = 0.0f;
}

// ---------------------------------------------------------------------------
// Kernel 1: scatter w_group[g,s] * fc[g] into wflat[group_idx[g,s]].
// Identical math to the reference gather (handles dup/missing indices too).
// ---------------------------------------------------------------------------
__global__ void tib_scatter_w(const int*   __restrict__ gidx,
                              const float* __restrict__ wg,
                              const float* __restrict__ fc,
                              float*       __restrict__ wflat) {
    int i = blockIdx.x * blockDim.x + threadIdx.x;
    if (i < G_GRP * S_GRP) {
        int g   = i / S_GRP;
        float v = wg[i] * fc[g];          // UNITS == 1
        atomicAdd(wflat + gidx[i], v);
    }
}

// ---------------------------------------------------------------------------
// Kernel 2: streaming GEMV out[b] = dot(x[b,:], wflat) via V_WMMA_F32_16X16X4.
//   A (16x4)  = wflat[kk..kk+3] broadcast to all 16 rows
//   B (4x16)  = x[rowBase+n, kk..kk+3] in column n
//   D[m][n]   = partial dot for row rowBase+n (same for all m)
// Lane L (n = L&15, h = L>>4) loads float2 at col kk+2h for both operands,
// matching the ISA 16x4 / 4x16 fp32 VGPR striping.
// Block = 512 threads = 16 waves; wave w owns k-slice [w*1024, (w+1)*1024).
// Grid = 256 blocks, one per 16-row tile. No divergence before the WMMAs.
// ---------------------------------------------------------------------------
__global__ void __launch_bounds__(512)
tib_gemv_wmma(const float* __restrict__ x,
              const float* __restrict__ wflat,
              float*       __restrict__ out) {
    const int lane    = threadIdx.x & 31;
    const int wave    = threadIdx.x >> 5;     // 0..15
    const int n       = lane & 15;
    const int h       = lane >> 4;            // 0 or 1
    const int rowBase = blockIdx.x << 4;      // 16 rows per block
    const int SLICE   = F_COLS / 16;          // 1024 columns per wave
    const int k0      = wave * SLICE;

    const float* xp = x + (size_t)(rowBase + n) * F_COLS + k0 + 2 * h;
    const float* wp = wflat + k0 + 2 * h;

#if __has_builtin(__builtin_amdgcn_wmma_f32_16x16x4_f32)
    v8f acc = {};
    #pragma unroll 8
    for (int kk = 0; kk < SLICE; kk += 4) {
        v2f bx = *(const v2f*)(xp + kk);      // global_load_b64: x[row, kk+2h..+1]
        v2f aw = *(const v2f*)(wp + kk);      // global_load_b64: wflat (L2-hot)
        acc = __builtin_amdgcn_wmma_f32_16x16x4_f32(
            /*neg_a=*/false, aw, /*neg_b=*/false, bx,
            /*c_mod=*/(short)0, acc, /*reuse_a=*/false, /*reuse_b=*/false);
    }
    // D is constant across m; lane n (<16) holds row rowBase+n in acc[0].
    if (lane < 16) atomicAdd(out + rowBase + n, acc[0]);
#else
    // Scalar fallback (keeps compile green if the f32 WMMA builtin is absent;
    // lanes h=0 and h=1 each cover half of every 4-col chunk, both atomicAdd).
    float r = 0.0f;
    #pragma unroll 8
    for (int kk = 0; kk < SLICE; kk += 4)
        r = fmaf(xp[kk], wp[kk], fmaf(xp[kk + 1], wp[kk + 1], r));
    atomicAdd(out + rowBase + n, r);
#endif
}

// ---------------------------------------------------------------------------
extern "C" void kernel_launch(void* const* d_in, const int* in_sizes, int n_in,
                              void* d_out, int out_size, void* d_ws, size_t ws_size,
                              hipStream_t stream) {
    const float* x    = (const float*)d_in[0];   // (B, F) fp32
    const int*   gidx = (const int*)d_in[1];     // (G, S) int
    const float* wg   = (const float*)d_in[2];   // (G, S) fp32
    const float* fc   = (const float*)d_in[3];   // (G, 1) fp32
    float*       out  = (float*)d_out;           // (B, 1) fp32
    float*       wflat = (float*)d_ws;           // F floats of scratch

    (void)in_sizes; (void)n_in; (void)out_size; (void)ws_size;

    tib_zero<<<F_COLS / 256, 256, 0, stream>>>(wflat, out);
    tib_scatter_w<<<(G_GRP * S_GRP) / 256, 256, 0, stream>>>(gidx, wg, fc, wflat);
    tib_gemv_wmma<<<B_ROWS / 16, 512, 0, stream>>>(x, wflat, out);
}